// SDNN_Network_13082470383702
// MI455X (gfx1250) — compile-verified
//
#include <hip/hip_runtime.h>

// ---------------------------------------------------------------------------
// Fully-fused SDNN forward for MI455X (gfx1250, wave32).
//   conv1(5x5 over T,H; 2->8) + bias + ReLU + LIF + delta
//   + conv_transpose(2x2 stride 2 over T,H; 8->2) + bias + ReLU
//   + 1x1 channel mix + LIF, all in one kernel; LIF state lives in VGPRs.
//
// conv1 runs on V_WMMA_F32_16X16X4_F32:
//   D[16x16] : M = 2*chan + tb (8 channels x 2 consecutive timesteps)
//              N = 16 consecutive w positions
//   K = 72   : k = dt'*12 + i, i = 2*dh + c (10 used of 12; stride 12 keeps
//              k mod 4 invariant under the t+=2 slide -> pure register slide)
//   A = conv1 weights (constant in VGPRs), B = sliding input window.
// D layout => lane (n, half) holds both timesteps of 4 channels in its own
// VGPRs, so the sequential LIF scan needs zero cross-lane traffic.
// B lane layout assumed to mirror A's documented K split:
//   VGPR0 = rows {4j (lanes 0-15), 4j+2 (lanes 16-31)}, VGPR1 = {4j+1,4j+3}.
//
// All loop trip counts / predicates are forced wave-uniform via
// readfirstlane so EXEC is provably all-ones at every WMMA (ISA req.)
// and the t-loop compiles as a scalar (non-divergent) loop.
// ---------------------------------------------------------------------------

typedef __attribute__((ext_vector_type(2))) float v2f;
typedef __attribute__((ext_vector_type(8))) float v8f;

namespace {
constexpr int   Bn = 2, Cin = 2, Hn = 64, Wn = 64, Tn = 350;
constexpr int   H2 = 2 * Hn, T2 = 2 * Tn;
constexpr int   NCHUNK = 5;             // T-chunks for occupancy
constexpr int   CHLEN  = Tn / NCHUNK;   // 70 (even)
constexpr int   WARM   = 10;            // LIF warm-up steps (0.05^10 ~ 1e-13)
constexpr int   WT     = Wn / 16;       // w tiles of 16 per wave
constexpr int   NWAVES = NCHUNK * Bn * Hn * WT;   // 2560 waves
constexpr float KEEP   = 0.05f;         // 1 - ALPHA
constexpr float TH     = 1.0f;
}

__global__ __launch_bounds__(256)
void sdnn_fused(const float* __restrict__ x,   const float* __restrict__ w1,
                const float* __restrict__ b1,  const float* __restrict__ dw,
                const float* __restrict__ dbv, const float* __restrict__ c2w,
                const float* __restrict__ c2b, float* __restrict__ out)
{
  const int tid  = threadIdx.x;
  const int lane = tid & 31;
  const int lh   = lane >> 4;   // lane half: B/D K-split & output-H parity
  const int sp   = lane & 15;   // w offset in tile; also A-matrix row (M)

  // Wave id, forced into an SGPR so everything derived from it is provably
  // wave-uniform (scalar loop bounds, scalar branches, full EXEC at WMMA).
  int wid = __builtin_amdgcn_readfirstlane(
      blockIdx.x * (blockDim.x >> 5) + (tid >> 5));
  const int wt = wid % WT; wid /= WT;
  const int h0 = wid % Hn; wid /= Hn;
  const int b  = wid % Bn; wid /= Bn;
  const int chunk = wid;
  const int w = wt * 16 + sp;

  // ---- A operands: conv1 weights, constant per lane (branchless) ----------
  // A chunk j, lane l: A[m = l%16, k = 4j + 2*lh + q], q in {0,1}
  // m = 2*chan + tb ; k = dt'*12 + i ; i = 2*dh + c ; kd = dt' - tb
  v2f aw[18];
#pragma unroll
  for (int j = 0; j < 18; ++j) {
    const int dtp = j / 3, p = j % 3;
    float v[2];
#pragma unroll
    for (int q = 0; q < 2; ++q) {
      const int i  = 4 * p + 2 * lh + q;
      const int tb = sp & 1, ch = sp >> 1;
      const int c  = i & 1,  dh = i >> 1;
      const int kd = dtp - tb;
      const int kdc = min(max(kd, 0), 4);
      const int dhc = min(dh, 4);
      const float wv = w1[((ch * Cin + c) * 5 + kdc) * 5 + dhc];
      v[q] = (dh < 5 && kd >= 0 && kd < 5) ? wv : 0.f;   // cndmask, no branch
    }
    aw[j] = v2f{v[0], v[1]};
  }

  // Accumulator seeded with conv1 bias: VGPR r holds (chan=(r>>1)+4*lh, tb=r&1)
  v8f cinit;
#pragma unroll
  for (int r = 0; r < 8; ++r) cinit[r] = b1[(r >> 1) + 4 * lh];

  // ---- B-operand address/mask tables (per lane, per (p,q) slot) -----------
  const long xb = (long)b * Cin * Hn * Wn * Tn;
  int   xoff[3][2];
  float xmask[3][2];
#pragma unroll
  for (int p = 0; p < 3; ++p)
#pragma unroll
    for (int q = 0; q < 2; ++q) {
      const int i = 4 * p + 2 * lh + q;
      const int c = i & 1, dh = i >> 1;
      const int hin = h0 + dh - 2;
      const bool ok = (dh < 5) && (hin >= 0) && (hin < Hn);
      const int hcl = min(max(hin, 0), Hn - 1);
      xoff[p][q]  = ((c * Hn + hcl) * Wn + w) * Tn;
      xmask[p][q] = ok ? 1.f : 0.f;
    }

  // ---- deconv weights for this lane's 4 channels ---------------------------
  // conv_transpose(stride2,VALID,transpose_kernel=False):
  //   out[o,2t+dtd,2h+dh] = sum_c dw[o,c,1-dtd,1-dh] * d[c,t,h]
  float dwv[2][4][2][2];  // [o][cc][dtd][dh]
#pragma unroll
  for (int o = 0; o < 2; ++o)
#pragma unroll
    for (int cc = 0; cc < 4; ++cc)
#pragma unroll
      for (int dtd = 0; dtd < 2; ++dtd)
#pragma unroll
        for (int dh = 0; dh < 2; ++dh)
          dwv[o][cc][dtd][dh] =
              dw[((o * 8 + (cc + 4 * lh)) * 2 + (1 - dtd)) * 2 + (1 - dh)];

  // small uniforms (wave-uniform -> SGPRs)
  const float db0 = dbv[0], db1 = dbv[1];
  const float c200 = c2w[0], c201 = c2w[1], c210 = c2w[2], c211 = c2w[3];
  const float cb0 = c2b[0], cb1 = c2b[1];

  // output column bases: lane half selects H' parity
  const int hp = 2 * h0 + lh;
  float* outb0 = out + (((long)(b * 2 + 0) * H2 + hp) * Wn + w) * (long)T2;
  float* outb1 = out + (((long)(b * 2 + 1) * H2 + hp) * Wn + w) * (long)T2;

  // ---- persistent recurrent state -----------------------------------------
  float y1[4] = {0, 0, 0, 0}, spv[4] = {0, 0, 0, 0};  // LIF1 + prev spike
  float y2a = 0.f, y2b = 0.f;                          // LIF2 (2 out channels)

  const int t0     = chunk * CHLEN;                    // SGPRs
  const int tstart = (chunk == 0) ? 0 : (t0 - WARM);
  const int niter  = (t0 + CHLEN - tstart) >> 1;

  auto loadcol = [&](int tt, v2f* col) {
    const int   tcl = min(max(tt, 0), Tn - 1);
    const float tm  = (tt >= 0 && tt < Tn) ? 1.f : 0.f;
#pragma unroll
    for (int p = 0; p < 3; ++p) {
      const float v0 = x[xb + xoff[p][0] + tcl] * (xmask[p][0] * tm);
      const float v1 = x[xb + xoff[p][1] + tcl] * (xmask[p][1] * tm);
      col[p] = v2f{v0, v1};
    }
  };

  // sliding window: xw2[dt'][p] covers input times t-2 .. t+3
  v2f xw2[6][3];
#pragma unroll
  for (int d = 0; d < 6; ++d) loadcol(tstart - 2 + d, xw2[d]);

  // unroll by the window-rotation period (3 pair-steps) so the slide copies
  // below become pure register renames after copy-propagation
#pragma unroll 3
  for (int it = 0; it < niter; ++it) {
    const int t = tstart + 2 * it;   // scalar

    // ---- conv1 via WMMA chain (K = 72) -----------------------------------
    v8f acc = cinit;
#pragma unroll
    for (int j = 0; j < 18; ++j) {
      acc = __builtin_amdgcn_wmma_f32_16x16x4_f32(
          false, aw[j], false, xw2[j / 3][j % 3], (short)0, acc, false, false);
    }

    // ---- ReLU + LIF1 + delta (all in-lane) -------------------------------
    float dd[2][4];  // [tb][cc] in {-1,0,1}
#pragma unroll
    for (int cc = 0; cc < 4; ++cc) {
      const float u0 = fmaxf(acc[2 * cc + 0], 0.f);
      const float u1 = fmaxf(acc[2 * cc + 1], 0.f);
      float y = y1[cc];
      y = KEEP * y + u0; const float s0 = (y >= TH) ? 1.f : 0.f; y *= (1.f - s0);
      y = KEEP * y + u1; const float s1 = (y >= TH) ? 1.f : 0.f; y *= (1.f - s1);
      y1[cc]   = y;
      dd[0][cc] = (t == 0) ? 0.f : (s0 - spv[cc]);  // delta_encode pads t=0
      dd[1][cc] = s1 - s0;
      spv[cc]  = s1;
    }

    const bool emit = (t >= t0);   // scalar predicate -> uniform branch
#pragma unroll
    for (int tb = 0; tb < 2; ++tb) {
      // deconv partial sums over this lane's 4 channels
      float pe[2][2][2] = {};  // [o][dtd][dh]
#pragma unroll
      for (int o = 0; o < 2; ++o)
#pragma unroll
        for (int cc = 0; cc < 4; ++cc)
#pragma unroll
          for (int dtd = 0; dtd < 2; ++dtd)
#pragma unroll
            for (int dh = 0; dh < 2; ++dh)
              pe[o][dtd][dh] += dwv[o][cc][dtd][dh] * dd[tb][cc];

      // half-wave reduce; each lane keeps its own dh (= lane half) row
      float E[2][2];
#pragma unroll
      for (int o = 0; o < 2; ++o)
#pragma unroll
        for (int dtd = 0; dtd < 2; ++dtd) {
          const float send = lh ? pe[o][dtd][0] : pe[o][dtd][1];
          const float keep = lh ? pe[o][dtd][1] : pe[o][dtd][0];
          E[o][dtd] = keep + __shfl_xor(send, 16, 32);
        }

      // bias + ReLU + 1x1 mix + LIF2 (t' = 2*(t+tb), 2*(t+tb)+1)
      float so[2][2];
#pragma unroll
      for (int dtd = 0; dtd < 2; ++dtd) {
        const float e0 = fmaxf(E[0][dtd] + db0, 0.f);
        const float e1 = fmaxf(E[1][dtd] + db1, 0.f);
        const float f0 = c200 * e0 + c201 * e1 + cb0;
        const float f1 = c210 * e0 + c211 * e1 + cb1;
        y2a = KEEP * y2a + f0; const float sa = (y2a >= TH) ? 1.f : 0.f; y2a *= (1.f - sa);
        y2b = KEEP * y2b + f1; const float sb = (y2b >= TH) ? 1.f : 0.f; y2b *= (1.f - sb);
        so[0][dtd] = sa; so[1][dtd] = sb;
      }
      if (emit) {  // wave-uniform
        const int tp = 2 * (t + tb);
        *(v2f*)(outb0 + tp) = v2f{so[0][0], so[0][1]};
        *(v2f*)(outb1 + tp) = v2f{so[1][0], so[1][1]};
      }
    }

    // ---- slide window by 2 timesteps (renamed away under unroll-3) -------
#pragma unroll
    for (int d = 0; d < 4; ++d) {
      xw2[d][0] = xw2[d + 2][0];
      xw2[d][1] = xw2[d + 2][1];
      xw2[d][2] = xw2[d + 2][2];
    }
    loadcol(t + 4, xw2[4]);
    loadcol(t + 5, xw2[5]);
  }
}

extern "C" void kernel_launch(void* const* d_in, const int* in_sizes, int n_in,
                              void* d_out, int out_size, void* d_ws, size_t ws_size,
                              hipStream_t stream) {
  (void)in_sizes; (void)n_in; (void)out_size; (void)d_ws; (void)ws_size;
  const float* x   = (const float*)d_in[0];
  const float* w1  = (const float*)d_in[1];
  const float* b1  = (const float*)d_in[2];
  const float* dwp = (const float*)d_in[3];
  const float* dbp = (const float*)d_in[4];
  const float* c2w = (const float*)d_in[5];
  const float* c2b = (const float*)d_in[6];
  float* outp = (float*)d_out;

  constexpr int threads = 256;                       // 8 wave32 per block
  constexpr int blocks  = NWAVES * 32 / threads;     // 2560 waves -> 320 blocks
  sdnn_fused<<<blocks, threads, 0, stream>>>(x, w1, b1, dwp, dbp, c2w, c2b, outp);
}